// SGC_CUSTOM_56435870269981
// MI455X (gfx1250) — compile-verified
//
#include <hip/hip_runtime.h>
#include <hip/hip_bf16.h>
#include <cstdint>

// ---------------------------------------------------------------------------
// SGConv K=2 on MI455X (gfx1250): edge-parallel scatter with
//  - TDM (tensor_load_to_lds) double-buffered staging of the packed edge
//    stream into LDS, pipelined with s_wait_tensorcnt (in-order per wave)
//  - wave-per-edge gather: 32 lanes x 3 b32 = one coalesced 384B source row
//  - non-returning global_atomic_add_f32 scatter (STOREcnt, no readback)
//  - global_prefetch_b8 on upcoming gathered rows
// Whole working set (~48MB) lives in the 192MB L2 after first touch.
// ---------------------------------------------------------------------------

#define FEAT 96
#define TILE_E 512              // edges per TDM tile
#define TILE_W (TILE_E * 3)     // dwords per tile {src,dst,norm}

#if defined(__HIP_DEVICE_COMPILE__) && defined(__gfx1250__) && \
    __has_builtin(__builtin_amdgcn_tensor_load_to_lds) &&       \
    __has_builtin(__builtin_amdgcn_s_wait_tensorcnt)
#define USE_TDM 1
#else
#define USE_TDM 0
#endif

#if USE_TDM
typedef unsigned int u32x4 __attribute__((ext_vector_type(4)));
typedef int          i32x8 __attribute__((ext_vector_type(8)));
typedef int          i32x4 __attribute__((ext_vector_type(4)));

// One 1-row 2D TDM tile: tile_elems dwords from gaddr -> LDS at lds_addr.
// tensor_dim0 = remaining dwords, so tail tiles zero-fill OOB reads (ISA 8.2).
__device__ __forceinline__ void tdm_issue_tile(const void* gaddr,
                                               unsigned lds_addr,
                                               unsigned tensor_elems,
                                               unsigned tile_elems) {
  unsigned long long ga = (unsigned long long)(uintptr_t)gaddr;
  u32x4 g0;
  g0.x = 1u;                                         // count=1, user descriptor
  g0.y = lds_addr;                                   // LDS byte address
  g0.z = (unsigned)(ga & 0xFFFFFFFFu);               // global_addr[31:0]
  g0.w = (unsigned)((ga >> 32) & 0x01FFFFFFu)        // global_addr[56:32]
       | (2u << 30);                                 // type = 2 ("image")
  i32x8 g1;
  g1[0] = (int)(2u << 16);                           // data_size = 4B
  g1[1] = (int)((tensor_elems & 0xFFFFu) << 16);     // tensor_dim0[15:0]
  g1[2] = (int)(((tensor_elems >> 16) & 0xFFFFu)     // tensor_dim0[31:16]
              | (1u << 16));                         // tensor_dim1 = 1
  g1[3] = (int)((tile_elems & 0xFFFFu) << 16);       // tile_dim0
  g1[4] = 1;                                         // tile_dim1=1, tile_dim2=0
  g1[5] = (int)tensor_elems;                         // tensor_dim0_stride lo
  g1[6] = 0;                                         // stride0 hi / stride1 lo
  g1[7] = 0;                                         // stride1 hi
  i32x4 z4 = {0, 0, 0, 0};                           // groups 2/3 unused (<=2D)
  i32x8 z8 = {0, 0, 0, 0, 0, 0, 0, 0};               // extension group unused
  __builtin_amdgcn_tensor_load_to_lds(g0, g1, z4, z4, z8, 0);
}
#endif

// --- degree / normalization prep --------------------------------------------
__global__ void k_deg_init(float* __restrict__ deg, int N) {
  int i = blockIdx.x * blockDim.x + threadIdx.x;
  if (i < N) deg[i] = 1.0f;  // self-loop weight
}

__global__ void k_deg_count(const int* __restrict__ ei, int E,
                            float* __restrict__ deg) {
  int i = blockIdx.x * blockDim.x + threadIdx.x;
  if (i < E) atomicAdd(&deg[ei[E + i]], 1.0f);  // row 1 = dst
}

__global__ void k_rsqrt(float* __restrict__ deg, int N) {
  int i = blockIdx.x * blockDim.x + threadIdx.x;
  if (i < N) deg[i] = 1.0f / sqrtf(deg[i]);  // deg >= 1 always
}

// pack {src, dst, norm} triples (one linear dword stream for the TDM)
__global__ void k_edge_pack(const int* __restrict__ ei, int E,
                            const float* __restrict__ dis,
                            int* __restrict__ er) {
  int i = blockIdx.x * blockDim.x + threadIdx.x;
  if (i < E) {
    int s = ei[i];
    int d = ei[E + i];
    er[3 * i + 0] = s;
    er[3 * i + 1] = d;
    float w = dis[s] * dis[d];
    er[3 * i + 2] = __float_as_int(w);
  }
}

// hout[n] = dis[n]^2 * hin[n]  (self-loop term; fully overwrites hout)
__global__ void k_self_init(const float4* __restrict__ hin,
                            const float* __restrict__ dis,
                            float4* __restrict__ hout, int n4) {
  int i = blockIdx.x * blockDim.x + threadIdx.x;
  if (i < n4) {
    int node = i / (FEAT / 4);
    float w = dis[node];
    w *= w;
    float4 v = hin[i];
    v.x *= w; v.y *= w; v.z *= w; v.w *= w;
    hout[i] = v;
  }
}

// --- edge scatter: hout[dst] += w * hin[src] --------------------------------
__global__ __launch_bounds__(256) void k_scatter(const int* __restrict__ er,
                                                 int E,
                                                 const float* __restrict__ hin,
                                                 float* __restrict__ hout) {
  __shared__ int sbuf[2][TILE_W];
  const int tid  = threadIdx.x;
  const int wid  = tid >> 5;   // 8 waves of 32 (wave32)
  const int lane = tid & 31;

  const int numTiles = (E + TILE_E - 1) / TILE_E;
  int t    = blockIdx.x;
  int step = gridDim.x;

#if USE_TDM
  if (t < numTiles && wid == 0) {
    tdm_issue_tile(er + 3 * t * TILE_E,
                   (unsigned)(unsigned long long)(uintptr_t)&sbuf[0][0],
                   (unsigned)(3 * (E - t * TILE_E)), TILE_W);
  }
#endif

  int buf = 0;
  for (; t < numTiles; t += step, buf ^= 1) {
    int tn = t + step;
#if USE_TDM
    if (wid == 0) {
      if (tn < numTiles) {
        tdm_issue_tile(er + 3 * tn * TILE_E,
                       (unsigned)(unsigned long long)(uintptr_t)&sbuf[buf ^ 1][0],
                       (unsigned)(3 * (E - tn * TILE_E)), TILE_W);
        __builtin_amdgcn_s_wait_tensorcnt(1);   // tile t retired (in-order)
      } else {
        __builtin_amdgcn_s_wait_tensorcnt(0);
      }
    }
    __syncthreads();
#else
    {
      long long base = 3LL * t * TILE_E;
      long long tot  = 3LL * E;
      for (int i = tid; i < TILE_W; i += 256)
        sbuf[buf][i] = (base + i < tot) ? er[base + i] : 0;
      __syncthreads();
    }
#endif

    const int* sb = sbuf[buf];
    int eBase = t * TILE_E;
    int cnt   = (E - eBase < TILE_E) ? (E - eBase) : TILE_E;

    for (int j = wid; j < cnt; j += 8) {      // 64 edges per wave per tile
      int   s = sb[3 * j + 0];                // LDS broadcast (same addr/wave)
      int   d = sb[3 * j + 1];
      float w = __int_as_float(sb[3 * j + 2]);

      int jn = j + 16;                        // prefetch a future source row
      if (jn < cnt) {
        int sn = sb[3 * jn];
        __builtin_prefetch(&hin[(size_t)sn * FEAT + lane], 0, 0);
      }

      const float* __restrict__ r = hin + (size_t)s * FEAT;
      float*       o              = hout + (size_t)d * FEAT;
      float a0 = r[lane];
      float a1 = r[lane + 32];
      float a2 = r[lane + 64];
      atomicAdd(&o[lane],      w * a0);       // global_atomic_add_f32, no rtn
      atomicAdd(&o[lane + 32], w * a1);
      atomicAdd(&o[lane + 64], w * a2);
    }
    __syncthreads();                          // buffer free for re-fill
  }
}

// ---------------------------------------------------------------------------
extern "C" void kernel_launch(void* const* d_in, const int* in_sizes, int n_in,
                              void* d_out, int out_size, void* d_ws,
                              size_t ws_size, hipStream_t stream) {
  const float* x  = (const float*)d_in[0];
  const int*   ei = (const int*)d_in[1];     // JAX default: int32, shape (2,E)
  float*       out = (float*)d_out;

  const int N = in_sizes[0] / FEAT;          // 50000
  const int E = in_sizes[1] / 2;             // 800000

  // workspace layout
  char*  ws   = (char*)d_ws;
  float* deg  = (float*)ws;                                    // N f32 (->dis)
  size_t o1   = ((size_t)N * 4 + 255) & ~(size_t)255;
  int*   er   = (int*)(ws + o1);                               // 3*E i32
  size_t o2   = o1 + (((size_t)E * 12 + 255) & ~(size_t)255);
  float* h1   = (float*)(ws + o2);                             // N*FEAT f32

  const dim3 blk(256);
  const int  n4 = N * (FEAT / 4);

  k_deg_init <<<(N + 255) / 256, blk, 0, stream>>>(deg, N);
  k_deg_count<<<(E + 255) / 256, blk, 0, stream>>>(ei, E, deg);
  k_rsqrt    <<<(N + 255) / 256, blk, 0, stream>>>(deg, N);
  k_edge_pack<<<(E + 255) / 256, blk, 0, stream>>>(ei, E, deg, er);

  // hop 1: h1 = A_hat * x
  k_self_init<<<(n4 + 255) / 256, blk, 0, stream>>>((const float4*)x, deg,
                                                    (float4*)h1, n4);
  k_scatter  <<<512, blk, 0, stream>>>(er, E, x, h1);

  // hop 2: out = A_hat * h1
  k_self_init<<<(n4 + 255) / 256, blk, 0, stream>>>((const float4*)h1, deg,
                                                    (float4*)out, n4);
  k_scatter  <<<512, blk, 0, stream>>>(er, E, h1, out);
}